// GaussianDiffusionTrainer_77034533421094
// MI455X (gfx1250) — compile-verified
//
#include <hip/hip_runtime.h>

// ---------------------------------------------------------------------------
// GaussianDiffusionTrainer forward-noise kernel for MI455X (gfx1250, wave32).
// out = x0 * P[t-1] + normal .* eye(32) * C[t-1]   over (B,3,32,32) fp32.
// Memory-bound (0.17 flop/byte): optimized for HBM streaming.
// gfx1250-specific paths used: wave32 ds-permute scans, non-temporal cache
// hints, and GLOBAL_LOAD_ASYNC_TO_LDS_B128 + s_wait_asynccnt pipelining.
// ---------------------------------------------------------------------------

typedef float v4f __attribute__((ext_vector_type(4)));
typedef int   v4i __attribute__((ext_vector_type(4)));

// exact parameter types of the async-LDS builtin (from hipcc diagnostics):
//   arg0: v4i in address space 1 (global, printed "__device__")
//   arg1: v4i in address space 3 (LDS,    printed "__shared__")
typedef __attribute__((address_space(1))) v4i g_v4i;
typedef __attribute__((address_space(3))) v4i l_v4i;

#define T_STEPS   1000
#define SCHED_N   (T_STEPS + 1)          // 1001 schedule entries (index 0..1000)
#define BETA_1f   1.0e-4f
#define BETA_Tf   0.02f

#if __has_builtin(__builtin_amdgcn_global_load_async_to_lds_b128)
#define HAVE_ASYNC_LDS 1
#else
#define HAVE_ASYNC_LDS 0
#endif

__device__ __forceinline__ void wait_async0() {
#if __has_builtin(__builtin_amdgcn_s_wait_asynccnt)
    __builtin_amdgcn_s_wait_asynccnt(0);
#else
    asm volatile("s_wait_asynccnt 0" ::: "memory");
#endif
    asm volatile("" ::: "memory");   // compiler ordering fence for the LDS read
}

__device__ __forceinline__ float beta_at(int k) {
    // jnp.linspace(BETA_1, BETA_T, T+1) in fp32: start + k * step
    const float step = (BETA_Tf - BETA_1f) / (float)T_STEPS;
    return BETA_1f + (float)k * step;
}

// ---------------------------------------------------------------------------
// One wave32 computes both schedule arrays:
//   ac[k] = cumprod(1-beta)        s[k] = sqrt(ac[k])
//   P[k]  = cumprod(s)             C[k] : c <- c*s[k] + beta[k]^2
// Each lane owns a 32-element chunk; cross-lane prefix via __shfl_up scans
// (product scans + non-commutative affine-composition scan for C).
// ---------------------------------------------------------------------------
__global__ __launch_bounds__(32) void schedule_kernel(float* __restrict__ P,
                                                      float* __restrict__ C) {
    const int lane = threadIdx.x;        // 0..31
    const int base = lane * 32;

    // ---- pass 1: local product of (1 - beta) over this lane's chunk -------
    float la = 1.0f;
    #pragma unroll
    for (int i = 0; i < 32; ++i) {
        int k = base + i;
        if (k < SCHED_N) la *= (1.0f - beta_at(k));
    }
    // inclusive product scan across lanes, then shift to exclusive
    float inc = la;
    #pragma unroll
    for (int off = 1; off < 32; off <<= 1) {
        float v = __shfl_up(inc, off, 32);
        if (lane >= off) inc *= v;
    }
    float acBase = __shfl_up(inc, 1, 32);
    if (lane == 0) acBase = 1.0f;

    // ---- pass 2: per-element ac, s; chunk-local P product and the
    //              chunk-local affine map (cA, cB) for the C recurrence -----
    float s[32];
    float ac = acBase;
    float ps = 1.0f;                 // local product of s
    float cA = 1.0f, cB = 0.0f;      // local affine: c_out = cA*c_in + cB
    #pragma unroll
    for (int i = 0; i < 32; ++i) {
        int k = base + i;
        float sv = 1.0f;
        if (k < SCHED_N) {
            float b = beta_at(k);
            ac *= (1.0f - b);
            sv  = __builtin_sqrtf(ac);
            ps *= sv;
            cB  = cB * sv + b * b;
            cA *= sv;
        }
        s[i] = sv;
    }

    // exclusive product scan of ps -> P base per lane
    float pinc = ps;
    #pragma unroll
    for (int off = 1; off < 32; off <<= 1) {
        float v = __shfl_up(pinc, off, 32);
        if (lane >= off) pinc *= v;
    }
    float pBase = __shfl_up(pinc, 1, 32);
    if (lane == 0) pBase = 1.0f;

    // exclusive affine-composition scan: T2∘T1 = (A2*A1, A2*B1 + B2)
    float A = cA, Bv = cB;
    #pragma unroll
    for (int off = 1; off < 32; off <<= 1) {
        float Ao = __shfl_up(A,  off, 32);
        float Bo = __shfl_up(Bv, off, 32);
        if (lane >= off) {
            Bv = A * Bo + Bv;   // use pre-update A (current applied last)
            A  = A * Ao;
        }
    }
    float cIn = __shfl_up(Bv, 1, 32);   // applied to initial c = 0
    if (lane == 0) cIn = 0.0f;

    // ---- pass 3: serial tail within chunk, write results ------------------
    float p = pBase;
    float c = cIn;
    #pragma unroll
    for (int i = 0; i < 32; ++i) {
        int k = base + i;
        if (k < SCHED_N) {
            float b = beta_at(k);
            p *= s[i];
            c  = c * s[i] + b * b;
            P[k] = p;
            C[k] = c;
        }
    }
}

// ---------------------------------------------------------------------------
// Main kernel: one 256-thread block per 32x32 channel-image (256 float4).
//   blockIdx.x == (b*3 + ch)  ->  b = blockIdx.x/3 is block-uniform, so the
//   timestep/coefficient fetches become s_load (scalar cache) ops.
// Software pipeline: async-copy the x0 quad to LDS (ASYNCcnt), overlap the
// scattered diagonal `normal` fetch, then wait + read back + FMA + NT store.
// ---------------------------------------------------------------------------
__global__ __launch_bounds__(256) void diffuse_kernel(
    const v4f*  __restrict__ x0,
    const float* __restrict__ nrm,
    const int*  __restrict__ tsteps,
    const float* __restrict__ P,
    const float* __restrict__ C,
    v4f* __restrict__ out) {

    __shared__ v4f tile[256];

    const int blk = blockIdx.x;          // channel-image index (b*3 + ch)
    const int tid = threadIdx.x;
    const int q   = blk * 256 + tid;     // global float4 index
    const int qcol = q & 7;              // quad column within row (0..7)
    const int row  = (q >> 3) & 31;      // image row i (0..31)

#if HAVE_ASYNC_LDS
    // stage 1: fire the bulk-stream copy into LDS (no VGPR destination).
    // Two-step casts: reinterpret in generic AS, then pure addrspacecast.
    v4i* gp = (v4i*)const_cast<v4f*>(&x0[q]);
    v4i* lp = (v4i*)&tile[tid];
    __builtin_amdgcn_global_load_async_to_lds_b128((g_v4i*)gp, (l_v4i*)lp, 0, 0);
#endif

    // stage 2 (overlapped): uniform coefficients + divergent diagonal fetch
    const int b   = blk / 3;             // uniform per block -> scalar loads
    const int t   = tsteps[b];           // t in [1, 1000]
    const float Pt = P[t - 1];
    const float Ct = C[t - 1];

    const bool onDiag = ((row >> 2) == qcol);  // diagonal elem in this quad?
    const int  d      = row & 3;               // component within quad
    float add = 0.0f;
    if (onDiag) {
        float nv = __builtin_nontemporal_load(&nrm[4 * q + d]);
        add = nv * Ct;
    }

    // stage 3: wait for the async copy, read back own slot (no barrier
    // needed: each lane reads only the LDS slot it wrote)
#if HAVE_ASYNC_LDS
    wait_async0();
    v4f x = tile[tid];
#else
    v4f x = __builtin_nontemporal_load(&x0[q]);
    (void)tile;
#endif

    v4f r;
    r.x = x.x * Pt;
    r.y = x.y * Pt;
    r.z = x.z * Pt;
    r.w = x.w * Pt;
    if (onDiag) {
        if      (d == 0) r.x += add;
        else if (d == 1) r.y += add;
        else if (d == 2) r.z += add;
        else             r.w += add;
    }

    __builtin_nontemporal_store(r, &out[q]);
}

extern "C" void kernel_launch(void* const* d_in, const int* in_sizes, int n_in,
                              void* d_out, int out_size, void* d_ws, size_t ws_size,
                              hipStream_t stream) {
    const v4f*   x0     = (const v4f*)d_in[0];
    const float* nrm    = (const float*)d_in[1];
    const int*   tsteps = (const int*)d_in[2];

    float* Pw = (float*)d_ws;        // 1001 floats
    float* Cw = Pw + 1024;           // 1001 floats (aligned slot)

    hipLaunchKernelGGL(schedule_kernel, dim3(1), dim3(32), 0, stream, Pw, Cw);

    const int n_images = in_sizes[0] / 1024;   // (B*3) channel-images
    hipLaunchKernelGGL(diffuse_kernel, dim3(n_images), dim3(256), 0, stream,
                       x0, nrm, tsteps, Pw, Cw, (v4f*)d_out);
}